// DiverseExpertLoss_RL_70068096466999
// MI455X (gfx1250) — compile-verified
//
#include <hip/hip_runtime.h>

#define BB 65536
#define CC 100
#define ALPHA 0.01f
#define TAU 2.0f
#define INV_TEMP 0.5f
#define EPSF 1e-9f
#define MSTRIDE 104
#define L2E 1.44269504088896340736f

typedef __attribute__((ext_vector_type(2))) float v2f;
typedef __attribute__((ext_vector_type(8))) float v8f;

// native exp2: v_exp_f32 (no libm, no extra mul)
__device__ __forceinline__ float fast_exp2(float x) { return __builtin_amdgcn_exp2f(x); }

// workspace float layout
#define WS_M    0        // 100*100 blended label matrix M
#define WS_LP   10000    // log_prior
#define WS_A3   10100    // lp - TAU*log(inv_prior+eps)
#define WS_U2   10200    // M @ lp
#define WS_U3   10300    // M @ a3
#define WS_CNT  10400    // int[100] bincount
#define WS_PART 10600    // 512*3 block partials (ce, kl, nsel)

__global__ void zero_counts_kernel(int* __restrict__ cnt) {
  int i = threadIdx.x;
  if (i < CC) cnt[i] = 0;
}

__global__ void hist_kernel(const int* __restrict__ target, int* __restrict__ cnt) {
  __shared__ int h[CC];
  if (threadIdx.x < CC) h[threadIdx.x] = 0;
  __syncthreads();
  int i = blockIdx.x * blockDim.x + threadIdx.x;
  int stride = gridDim.x * blockDim.x;
  for (; i < BB; i += stride) atomicAdd(&h[target[i]], 1);
  __syncthreads();
  if (threadIdx.x < CC) atomicAdd(&cnt[threadIdx.x], h[threadIdx.x]);
}

// One block: log_prior, inverse_prior (stable-rank mirror), M = (1-a)I + a*softmax(cos),
// u2 = M@lp, u3 = M@a3.
__global__ void precompute_kernel(const float* __restrict__ cosf,
                                  const float* __restrict__ prior,
                                  float* __restrict__ ws) {
  __shared__ float pr[CC];
  __shared__ int   rank[CC];
  __shared__ float lp[CC], a3[CC];
  int t = threadIdx.x;
  if (t < CC) pr[t] = prior[t];
  __syncthreads();
  if (t < CC) {
    int r = 0; float b = pr[t];
    for (int j = 0; j < CC; ++j) {
      float a = pr[j];
      if (a < b || (a == b && j < t)) r++;
    }
    rank[t] = r;
  }
  __syncthreads();
  if (t < CC) {
    int want = CC - 1 - rank[t];      // mirrored rank
    float v = 0.f;
    for (int j = 0; j < CC; ++j) if (rank[j] == want) v = pr[j];
    float l = __logf(pr[t] + EPSF);
    lp[t] = l;
    a3[t] = l - TAU * __logf(v + EPSF);
    ws[WS_LP + t] = l;
    ws[WS_A3 + t] = a3[t];
  }
  __syncthreads();
  if (t < CC) {
    const float* row = cosf + t * CC;
    float mx = -__builtin_inff();
    for (int c = 0; c < CC; ++c) mx = fmaxf(mx, row[c]);
    float s = 0.f;
    for (int c = 0; c < CC; ++c) s += __expf(row[c] - mx);
    float inv = 1.f / s;
    float u2 = 0.f, u3 = 0.f;
    for (int c = 0; c < CC; ++c) {
      float m = ALPHA * __expf(row[c] - mx) * inv + ((c == t) ? (1.f - ALPHA) : 0.f);
      ws[WS_M + t * CC + c] = m;
      u2 += m * lp[c];
      u3 += m * a3[c];
    }
    ws[WS_U2 + t] = u2;
    ws[WS_U3 + t] = u3;
  }
}

// Fused main kernel: 256 threads = 8 waves; each wave owns a 16-row tile.
// Lane pair (L, L+16) owns row L: L -> classes c%4 in {0,1}, L+16 -> {2,3}.
// This matches the V_WMMA_F32_16X16X4_F32 A/B operand layouts exactly.
__global__ __launch_bounds__(256) void main_kernel(
    const int*   __restrict__ target,
    const float* __restrict__ old_pred,
    const float* __restrict__ e1g,
    const float* __restrict__ e2g,
    const float* __restrict__ e3g,
    const float* __restrict__ outp,
    const float* __restrict__ ws,
    const int*   __restrict__ cnt,
    float*       __restrict__ es_out,     // d_out + 2
    float*       __restrict__ partials) { // [512*3]
  __shared__ __align__(16) float Ml[CC * MSTRIDE];
  __shared__ __align__(16) float lp_s[MSTRIDE], a3_s[MSTRIDE];
  __shared__ float u2_s[CC], u3_s[CC];
  __shared__ float wsum[8][4];

  int tid = threadIdx.x;
  for (int i = tid; i < CC * CC; i += 256) {
    int r = i / CC, c = i - r * CC;
    Ml[r * MSTRIDE + c] = ws[WS_M + i];
  }
  if (tid < CC) {
    lp_s[tid] = ws[WS_LP + tid];
    a3_s[tid] = ws[WS_A3 + tid];
    u2_s[tid] = ws[WS_U2 + tid];
    u3_s[tid] = ws[WS_U3 + tid];
  }
  __syncthreads();

  int wave = tid >> 5, lane = tid & 31;
  int m = lane & 15, half = lane >> 4;
  int row = (blockIdx.x * 8 + wave) * 16 + m;
  int t = target[row];
  float inv_num = 1.f / (float)cnt[t];

  const float* r1 = e1g + row * CC + 2 * half;
  const float* r2 = e2g + row * CC + 2 * half;
  const float* r3 = e3g + row * CC + 2 * half;
  const float* rt = old_pred + row * CC + 2 * half;
  const float* ro = outp + row * CC + 2 * half;

  v8f acc1 = {}, acc2 = {}, acc3 = {};
  float mx1 = -__builtin_inff(), s1 = 0.f;
  float mx2 = -__builtin_inff(), s2 = 0.f;
  float mx3 = -__builtin_inff(), s3 = 0.f;
  float mo = -__builtin_inff();            // raw student max
  float bv = -__builtin_inff(); int bc = CC; // teacher argmax (raw)

  // 1-exp online logsumexp update: one of exp(mx-nm), exp(x-nm) is always 1.
  // Branchless: running max reuses the d>0 compare (no fmax canonicalize ops).
  auto upd = [](float& mx, float& s, float x) {
    float d = x - mx;
    float e = fast_exp2(-fabsf(d) * L2E);
    bool up = d > 0.f;
    s = fmaf(s, up ? e : 1.f, up ? 1.f : e);
    mx = up ? x : mx;
  };

  #pragma unroll 5
  for (int j = 0; j < 25; ++j) {
    int c0 = 4 * j + 2 * half;
    v2f a1 = *(const v2f*)(r1 + 4 * j);
    v2f a2 = *(const v2f*)(r2 + 4 * j);
    v2f a3v = *(const v2f*)(r3 + 4 * j);
    v2f xt = *(const v2f*)(rt + 4 * j);
    v2f xo = *(const v2f*)(ro + 4 * j);
    v2f bvec = *(const v2f*)(&Ml[t * MSTRIDE + c0]);
    v2f lpv = *(const v2f*)(&lp_s[c0]);
    v2f adv = *(const v2f*)(&a3_s[c0]);

    // dots: diag(A x B), A = 16x4 logits tile, B[k][n] = M[target_n][k]
    acc1 = __builtin_amdgcn_wmma_f32_16x16x4_f32(false, a1,  false, bvec, (short)0, acc1, false, false);
    acc2 = __builtin_amdgcn_wmma_f32_16x16x4_f32(false, a2,  false, bvec, (short)0, acc2, false, false);
    acc3 = __builtin_amdgcn_wmma_f32_16x16x4_f32(false, a3v, false, bvec, (short)0, acc3, false, false);

    #pragma unroll
    for (int k = 0; k < 2; ++k) {
      upd(mx1, s1, a1[k]);
      upd(mx2, s2, a2[k] + lpv[k]);
      upd(mx3, s3, a3v[k] + adv[k]);
      mo = (xo[k] > mo) ? xo[k] : mo;
      int c = c0 + k;
      float xv = xt[k];
      bool better = (xv > bv) || ((xv == bv) && (c < bc));  // branchless argmax
      bv = better ? xv : bv;
      bc = better ? c : bc;
    }
  }

  // combine lane pair (xor 16) -> logsumexps
  auto pairlse = [&](float mx, float s) -> float {
    float mxo = __shfl_xor(mx, 16, 32);
    float so = __shfl_xor(s, 16, 32);
    float nm = fmaxf(mx, mxo);
    float sc = s * fast_exp2((mx - nm) * L2E) + so * fast_exp2((mxo - nm) * L2E);
    return nm + __logf(sc);
  };
  float lse1 = pairlse(mx1, s1), lse2 = pairlse(mx2, s2), lse3 = pairlse(mx3, s3);

  {
    float bvo = __shfl_xor(bv, 16, 32);
    int   bco = __shfl_xor(bc, 16, 32);
    bool better = (bvo > bv) || ((bvo == bv) && (bco < bc));
    bv = better ? bvo : bv;
    bc = better ? bco : bc;
  }
  float mt = bv * INV_TEMP;                                   // teacher scaled max
  float ms = fmaxf(mo, __shfl_xor(mo, 16, 32)) * inv_num * INV_TEMP; // student scaled max

  // diag extraction: lanes 0-7 hold diag(m=lane) at acc[lane];
  // lanes 24-31 hold diag(m=lane-16) at acc[lane-24]
  auto diag = [&](v8f a) -> float {
    float d = 0.f;
    int sel = lane & 7;
    #pragma unroll
    for (int v = 0; v < 8; ++v) d = (sel == v) ? a[v] : d;
    int src = (m < 8) ? m : (m + 16);
    return __shfl(d, src, 32);
  };
  float d1 = diag(acc1), d2 = diag(acc2), d3 = diag(acc3);

  float ce_row = (lse1 - d1) + (lse2 - d2 - u2_s[t]) + (lse3 - d3 - u3_s[t]);

  // pass 2 (cache-hot): teacher/student exp-sums vs known maxes, KL pieces, es store
  float st2 = 0.f, aT = 0.f, aS = 0.f, ss2 = 0.f;
  #pragma unroll 5
  for (int j = 0; j < 25; ++j) {
    int c0 = 4 * j + 2 * half;
    v2f xt = *(const v2f*)(rt + 4 * j);
    v2f xo = *(const v2f*)(ro + 4 * j);
    v2f es;
    es[0] = xo[0] * inv_num;
    es[1] = xo[1] * inv_num;
    *(v2f*)(es_out + row * CC + c0) = es;
    #pragma unroll
    for (int k = 0; k < 2; ++k) {
      float xts = xt[k] * INV_TEMP;
      float et = fast_exp2((xts - mt) * L2E);
      st2 += et;
      aT = fmaf(et, xts, aT);
      float xss = es[k] * INV_TEMP;
      aS = fmaf(et, xss, aS);
      ss2 += fast_exp2((xss - ms) * L2E);
    }
  }
  st2 += __shfl_xor(st2, 16, 32);
  aT  += __shfl_xor(aT, 16, 32);
  aS  += __shfl_xor(aS, 16, 32);
  ss2 += __shfl_xor(ss2, 16, 32);
  float lse_t = mt + __logf(st2);
  float lse_s = ms + __logf(ss2);
  float kl_row = (aT - aS) / st2 - lse_t + lse_s;
  float maskf = (bc == t) ? 1.f : 0.f;

  // wave reduction (dedupe pair: lanes 0-15 only), fixed order
  float vce = (lane < 16) ? ce_row : 0.f;
  float vkl = (lane < 16) ? kl_row * maskf : 0.f;
  float vns = (lane < 16) ? maskf : 0.f;
  #pragma unroll
  for (int off = 16; off >= 1; off >>= 1) {
    vce += __shfl_xor(vce, off, 32);
    vkl += __shfl_xor(vkl, off, 32);
    vns += __shfl_xor(vns, off, 32);
  }
  if (lane == 0) { wsum[wave][0] = vce; wsum[wave][1] = vkl; wsum[wave][2] = vns; }
  __syncthreads();
  if (tid == 0) {
    float L = 0.f, K = 0.f, N = 0.f;
    for (int w = 0; w < 8; ++w) { L += wsum[w][0]; K += wsum[w][1]; N += wsum[w][2]; }
    partials[blockIdx.x * 3 + 0] = L;
    partials[blockIdx.x * 3 + 1] = K;
    partials[blockIdx.x * 3 + 2] = N;
  }
}

__global__ void finalize_kernel(const float* __restrict__ partials, float* __restrict__ out) {
  if (threadIdx.x == 0) {
    float L = 0.f, K = 0.f, N = 0.f;
    for (int b = 0; b < 512; ++b) {
      L += partials[3 * b + 0];
      K += partials[3 * b + 1];
      N += partials[3 * b + 2];
    }
    out[0] = L / (float)BB;
    out[1] = (N > 0.f) ? (K / fmaxf(N, 1.f)) * 12.f : 0.f;  // TEMP^2 * 3
  }
}

extern "C" void kernel_launch(void* const* d_in, const int* in_sizes, int n_in,
                              void* d_out, int out_size, void* d_ws, size_t ws_size,
                              hipStream_t stream) {
  (void)in_sizes; (void)n_in; (void)out_size; (void)ws_size;
  // inputs: 0 output_logits(unused), 1 target, 2 cos_feature, 3 old_pred,
  //         4 expert1, 5 expert2, 6 expert3, 7 output, 8 prior, 9 epoch(unused)
  const int*   target   = (const int*)d_in[1];
  const float* cosf     = (const float*)d_in[2];
  const float* old_pred = (const float*)d_in[3];
  const float* e1       = (const float*)d_in[4];
  const float* e2       = (const float*)d_in[5];
  const float* e3       = (const float*)d_in[6];
  const float* outp     = (const float*)d_in[7];
  const float* prior    = (const float*)d_in[8];

  float* wsf = (float*)d_ws;
  int* cnt = (int*)(wsf + WS_CNT);
  float* partials = wsf + WS_PART;
  float* outf = (float*)d_out;

  zero_counts_kernel<<<1, 128, 0, stream>>>(cnt);
  hist_kernel<<<256, 256, 0, stream>>>(target, cnt);
  precompute_kernel<<<1, 128, 0, stream>>>(cosf, prior, wsf);
  main_kernel<<<512, 256, 0, stream>>>(target, old_pred, e1, e2, e3, outp,
                                       wsf, cnt, outf + 2, partials);
  finalize_kernel<<<1, 32, 0, stream>>>(partials, outf);
}